// HiPPO_LegT_54778012893373
// MI455X (gfx1250) — compile-verified
//
#include <hip/hip_runtime.h>
#include <hip/hip_bf16.h>

// HiPPO-LegT scan for MI455X (gfx1250), chunked linear-scan formulation.
// All matrix math via V_WMMA_F32_16X16X4_F32 (fp32 required: 1024-step
// recurrence with spectral radius ~1; kernel is HBM-store bound anyway).

typedef float v2f __attribute__((ext_vector_type(2)));
typedef float v8f __attribute__((ext_vector_type(8)));

#define NDIM   256           // state dimension N
#define LTOT   1024          // sequence length
#define RTOT   512           // rows = batch(8) * M(64)
#define CHUNK  64            // steps per chunk
#define NCHUNK 16            // LTOT / CHUNK
#define STRIPS 32            // RTOT / 16 row-strips
#define AT_LD  264           // LDS row stride (floats) for A^T: 2*AT_LD % 64 == 16
                             //  -> half-wave K-slices land on disjoint bank groups
#define C_LD   17            // LDS stride for state stored [k][m]; 17 coprime 64

#define LDS_FLOATS (NDIM*AT_LD + 2*NDIM*C_LD + 16*CHUNK)
#define LDS_BYTES  (LDS_FLOATS * 4)

static __device__ __forceinline__ v8f wmma_f32(v2f a, v2f b, v8f c) {
  // D = A(16x4) * B(4x16) + C(16x16), fp32
  return __builtin_amdgcn_wmma_f32_16x16x4_f32(
      /*neg_a=*/false, a, /*neg_b=*/false, b,
      /*c_mod=*/(short)0, c, /*reuse_a=*/false, /*reuse_b=*/false);
}

// ---------------------------------------------------------------- transpose
__global__ __launch_bounds__(256) void k_transpose(const float* __restrict__ A,
                                                   float* __restrict__ At) {
  int i = blockIdx.x * 256 + threadIdx.x;   // 65536 threads
  int k = i >> 8, n = i & 255;
  At[k * NDIM + n] = A[n * NDIM + k];       // At[k][n] = A[n][k]
}

// ---------------------------------------------------------------- zero fill
__global__ __launch_bounds__(256) void k_zero(float* __restrict__ p, int n) {
  int i = blockIdx.x * 256 + threadIdx.x;
  if (i < n) p[i] = 0.0f;
}

// ------------------------------------------------ generic 16x16-tile WMMA GEMM
// Dst[M x 256] = Aop[M x 256] * Bop[256 x 256] (+ Cin), row-major fp32.
// One wave per 16x16 output tile; grid.x = (M/16)*16 tiles.
__global__ __launch_bounds__(32)
void k_gemm16(const float* __restrict__ Aop, const float* __restrict__ Bop,
              const float* __restrict__ Cin, float* __restrict__ Dst) {
  int mt = blockIdx.x >> 4;     // row tile
  int nt = blockIdx.x & 15;     // col tile
  int lane = threadIdx.x & 31;
  int lo = lane & 15, hi = lane >> 4;

  v8f acc = {0.f, 0.f, 0.f, 0.f, 0.f, 0.f, 0.f, 0.f};
  if (Cin) {
#pragma unroll
    for (int v = 0; v < 8; ++v)
      acc[v] = Cin[(size_t)(mt * 16 + v + 8 * hi) * NDIM + nt * 16 + lo];
  }
  const float* arow = Aop + (size_t)(mt * 16 + lo) * NDIM;  // lane's M-row
  const float* bcol = Bop + nt * 16 + lo;                   // lane's N-col
#pragma unroll 4
  for (int k0 = 0; k0 < NDIM; k0 += 4) {
    int ka = k0 + 2 * hi;
    v2f a, b;
    a.x = arow[ka];
    a.y = arow[ka + 1];
    b.x = bcol[(size_t)ka * NDIM];
    b.y = bcol[(size_t)(ka + 1) * NDIM];
    acc = wmma_f32(a, b, acc);
  }
#pragma unroll
  for (int v = 0; v < 8; ++v)
    Dst[(size_t)(mt * 16 + v + 8 * hi) * NDIM + nt * 16 + lo] = acc[v];
}

// ------------------------------------------------------- chunk recurrence
// mode 0: run chunk from zero state, emit end-state E[chunk] only.
// mode 1: run chunk seeded with S[chunk], emit all per-step outputs.
// Block = 256 threads (8 waves); wave w owns N-columns [32w, 32w+32).
// A^T (256x256 fp32) fully LDS-resident (270 KB of the 320 KB WGP LDS).
__global__ __launch_bounds__(256)
void k_chunk(const float* __restrict__ f,     // (512, 1024) row-major (r, t)
             const float* __restrict__ At_g,  // (256, 256) = A^T
             const float* __restrict__ Bvec,  // (256,)
             const float* __restrict__ S,     // (16, 512, 256) chunk seeds
             float* __restrict__ E,           // (16, 512, 256) chunk end-states
             float* __restrict__ out,         // (1024, 512, 256)
             int mode) {
  extern __shared__ float lds[];
  float* at_l = lds;                       // [256][AT_LD]
  float* c_l  = at_l + NDIM * AT_LD;       // 2 x [256][C_LD], state stored [k][m]
  float* f_l  = c_l + 2 * NDIM * C_LD;     // [16][CHUNK]

  const int tid = threadIdx.x;
  const int wave = tid >> 5;
  const int lane = tid & 31;
  const int lo = lane & 15, hi = lane >> 4;
  const int strip = blockIdx.x;            // 0..31 (16-row strip)
  const int chunk = blockIdx.y;            // 0..15
  const int r0 = strip * 16;
  const int t0 = chunk * CHUNK;

  // Stage A^T into LDS (padded rows).
  for (int i = tid; i < NDIM * NDIM; i += 256) {
    int k = i >> 8, n = i & 255;
    at_l[k * AT_LD + n] = At_g[i];
  }
  // Stage this strip's chunk of f: f_l[m][j] = f[r0+m, t0+j].
  for (int i = tid; i < 16 * CHUNK; i += 256) {
    int m = i >> 6, j = i & (CHUNK - 1);
    f_l[m * CHUNK + j] = f[(size_t)(r0 + m) * LTOT + t0 + j];
  }
  // Seed state buffer 0 (stored transposed: c_l[k][m]).
  const float* Sblk = S + ((size_t)chunk * RTOT + r0) * NDIM;
  for (int i = tid; i < 16 * NDIM; i += 256) {
    int m = i >> 8, n = i & 255;
    c_l[n * C_LD + m] = mode ? Sblk[(size_t)m * NDIM + n] : 0.0f;
  }
  __syncthreads();

  const int n0 = wave * 32;
  const float bv0 = Bvec[n0 + lo];
  const float bv1 = Bvec[n0 + 16 + lo];

  for (int j = 0; j < CHUNK; ++j) {
    const float* cprev = c_l + (j & 1) * (NDIM * C_LD);
    float*       cnext = c_l + ((j + 1) & 1) * (NDIM * C_LD);

    v8f acc0 = {0.f, 0.f, 0.f, 0.f, 0.f, 0.f, 0.f, 0.f};
    v8f acc1 = {0.f, 0.f, 0.f, 0.f, 0.f, 0.f, 0.f, 0.f};
#pragma unroll 8
    for (int k0 = 0; k0 < NDIM; k0 += 4) {
      int ka = k0 + 2 * hi;
      v2f a, b0, b1;
      a.x  = cprev[ka * C_LD + lo];           // c[m=lo, ka]
      a.y  = cprev[(ka + 1) * C_LD + lo];
      b0.x = at_l[ka * AT_LD + n0 + lo];      // At[ka, n]
      b0.y = at_l[(ka + 1) * AT_LD + n0 + lo];
      b1.x = at_l[ka * AT_LD + n0 + 16 + lo];
      b1.y = at_l[(ka + 1) * AT_LD + n0 + 16 + lo];
      acc0 = wmma_f32(a, b0, acc0);
      acc1 = wmma_f32(a, b1, acc1);
    }
    // D += f_t (x) B ; publish new state (and outputs in mode 1).
#pragma unroll
    for (int v = 0; v < 8; ++v) {
      int m = v + 8 * hi;
      float fv = f_l[m * CHUNK + j];          // broadcast within half-wave
      float d0 = acc0[v] + fv * bv0;
      float d1 = acc1[v] + fv * bv1;
      cnext[(n0 + lo) * C_LD + m]      = d0;
      cnext[(n0 + 16 + lo) * C_LD + m] = d1;
      if (mode) {
        size_t ob = ((size_t)(t0 + j) * RTOT + r0 + m) * NDIM;
        out[ob + n0 + lo]      = d0;
        out[ob + n0 + 16 + lo] = d1;
      }
    }
    __syncthreads();
  }

  if (!mode) {  // end state lives in buffer (CHUNK & 1) == 0
    const float* cfin = c_l;
    for (int i = tid; i < 16 * NDIM; i += 256) {
      int m = i >> 8, n = i & 255;
      E[((size_t)chunk * RTOT + r0 + m) * NDIM + n] = cfin[n * C_LD + m];
    }
  }
}

// ---------------------------------------------------------------- launcher
extern "C" void kernel_launch(void* const* d_in, const int* in_sizes, int n_in,
                              void* d_out, int out_size, void* d_ws, size_t ws_size,
                              hipStream_t stream) {
  (void)in_sizes; (void)n_in; (void)out_size; (void)ws_size;
  const float* f  = (const float*)d_in[0];   // (8,64,1024)
  const float* A  = (const float*)d_in[1];   // (256,256)
  const float* Bv = (const float*)d_in[2];   // (256,)
  float* out = (float*)d_out;                // (1024,512,256)

  float* ws = (float*)d_ws;                  // ~16.8 MB of fp32 scratch
  float* At = ws;                            // 65536
  float* T0 = ws + 65536;                    // 65536
  float* T1 = ws + 131072;                   // 65536
  float* E  = ws + 196608;                   // 16*512*256
  float* S  = E + (size_t)NCHUNK * RTOT * NDIM;

  hipFuncSetAttribute((const void*)k_chunk,
                      hipFuncAttributeMaxDynamicSharedMemorySize, LDS_BYTES);

  // 1) A^T, and S_0 = 0
  k_transpose<<<256, 256, 0, stream>>>(A, At);
  k_zero<<<512, 256, 0, stream>>>(S, RTOT * NDIM);

  // 2) P = (A^T)^64 by repeated squaring (ends in T1)
  k_gemm16<<<256, 32, 0, stream>>>(At, At, nullptr, T0);   // ^2
  k_gemm16<<<256, 32, 0, stream>>>(T0, T0, nullptr, T1);   // ^4
  k_gemm16<<<256, 32, 0, stream>>>(T1, T1, nullptr, T0);   // ^8
  k_gemm16<<<256, 32, 0, stream>>>(T0, T0, nullptr, T1);   // ^16
  k_gemm16<<<256, 32, 0, stream>>>(T1, T1, nullptr, T0);   // ^32
  k_gemm16<<<256, 32, 0, stream>>>(T0, T0, nullptr, T1);   // ^64

  // 3) all chunks from zero state -> E[k]  (parallel over 16 x 32 blocks)
  dim3 grid(STRIPS, NCHUNK);
  k_chunk<<<grid, 256, LDS_BYTES, stream>>>(f, At, Bv, S, E, out, 0);

  // 4) boundary scan: S_k = S_{k-1} * P + E_{k-1}   (15 short GEMMs)
  for (int k = 1; k < NCHUNK; ++k)
    k_gemm16<<<512, 32, 0, stream>>>(S + (size_t)(k - 1) * RTOT * NDIM, T1,
                                     E + (size_t)(k - 1) * RTOT * NDIM,
                                     S + (size_t)k * RTOT * NDIM);

  // 5) re-run chunks seeded with S_k, streaming out all 1024 states
  k_chunk<<<grid, 256, LDS_BYTES, stream>>>(f, At, Bv, S, E, out, 1);
}